// CrossConv2d_79594333929496
// MI455X (gfx1250) — compile-verified
//
#include <hip/hip_runtime.h>

typedef __attribute__((ext_vector_type(2))) float v2f;
typedef __attribute__((ext_vector_type(4))) float f4;
typedef __attribute__((ext_vector_type(8))) float v8f;
typedef __attribute__((ext_vector_type(4))) int   v4i;

#define HW 16384   // 128*128
#define WID 128
#define EPSV 1e-5f

// ---------------- CDNA5 async global->LDS copy (ASYNCcnt path) -------------
#if defined(__gfx1250__) && __has_builtin(__builtin_amdgcn_global_load_async_to_lds_b128)
#define HAVE_ASYNC_LDS 1
#else
#define HAVE_ASYNC_LDS 0
#endif

__device__ __forceinline__ void copy16_g2l(const float* g, float* l) {
#if HAVE_ASYNC_LDS
    // signature (from clang diagnostic): (v4i __device__*, v4i __shared__*, Ii, Ii)
    __builtin_amdgcn_global_load_async_to_lds_b128(
        (__attribute__((address_space(1))) v4i*)const_cast<float*>(g),
        (__attribute__((address_space(3))) v4i*)l,
        0, 0);
#else
    *(f4*)l = *(const f4*)g;
#endif
}

__device__ __forceinline__ void async_wait_all() {
#if HAVE_ASYNC_LDS
#if __has_builtin(__builtin_amdgcn_s_wait_asynccnt)
    __builtin_amdgcn_s_wait_asynccnt(0);
#else
    asm volatile("s_wait_asynccnt 0" ::: "memory");
#endif
#endif
}

// ---------------------------------------------------------------------------
// Kernel 1: fused 1x1 conv + BN + ReLU as GEMM on V_WMMA_F32_16X16X4_F32
// grid = (128 pixel-tiles of 128 px, 8 images), block = 256 (8 waves)
// each wave: 64 oc x 16 px (4 accumulators), K = 64 in steps of 4
// Input is read exactly once -> non-temporal loads (preserve L2 for ws).
// ---------------------------------------------------------------------------
__global__ __launch_bounds__(256)
void proj_bn_relu_kernel(const float* __restrict__ x,      // (8,64,128,128)
                         const float* __restrict__ w,      // (64,64)
                         const float* __restrict__ gamma,
                         const float* __restrict__ beta,
                         const float* __restrict__ mean,
                         const float* __restrict__ var,
                         float* __restrict__ out)          // (8,64,128,128)
{
    __shared__ float sW[64 * 64];
    __shared__ float sScale[64];
    __shared__ float sShift[64];

    const int tid = threadIdx.x;
    for (int i = tid; i < 64 * 64; i += 256) sW[i] = w[i];
    if (tid < 64) {
        float s = gamma[tid] / sqrtf(var[tid] + EPSV);
        sScale[tid] = s;
        sShift[tid] = beta[tid] - mean[tid] * s;
    }
    __syncthreads();

    const int wave = tid >> 5;
    const int lane = tid & 31;
    const int ln   = lane & 15;   // N index within half
    const int hi   = lane >> 4;   // lane-half: K split for A/B, M+8 for C/D

    const int img   = blockIdx.y;
    const int p     = blockIdx.x * 128 + wave * 16 + ln;
    const float* xi = x + (size_t)img * 64 * HW;

    v8f acc[4] = {v8f{}, v8f{}, v8f{}, v8f{}};

    for (int k0 = 0; k0 < 64; k0 += 4) {
        const int kr = k0 + hi * 2;
        v2f b;
        b.x = __builtin_nontemporal_load(xi + (size_t)kr * HW + p);
        b.y = __builtin_nontemporal_load(xi + (size_t)(kr + 1) * HW + p);
        v2f a0 = *(const v2f*)&sW[( 0 + ln) * 64 + k0 + hi * 2];
        v2f a1 = *(const v2f*)&sW[(16 + ln) * 64 + k0 + hi * 2];
        v2f a2 = *(const v2f*)&sW[(32 + ln) * 64 + k0 + hi * 2];
        v2f a3 = *(const v2f*)&sW[(48 + ln) * 64 + k0 + hi * 2];
        acc[0] = __builtin_amdgcn_wmma_f32_16x16x4_f32(false, a0, false, b, (short)0, acc[0], false, false);
        acc[1] = __builtin_amdgcn_wmma_f32_16x16x4_f32(false, a1, false, b, (short)0, acc[1], false, false);
        acc[2] = __builtin_amdgcn_wmma_f32_16x16x4_f32(false, a2, false, b, (short)0, acc[2], false, false);
        acc[3] = __builtin_amdgcn_wmma_f32_16x16x4_f32(false, a3, false, b, (short)0, acc[3], false, false);
    }

    float* oi = out + (size_t)img * 64 * HW;
    #pragma unroll
    for (int m = 0; m < 4; ++m) {
        #pragma unroll
        for (int r = 0; r < 8; ++r) {
            const int oc = m * 16 + hi * 8 + r;
            float vv = acc[m][r] * sScale[oc] + sShift[oc];
            vv = fmaxf(vv, 0.0f);
            oi[(size_t)oc * HW + p] = vv;
        }
    }
}

// ---------------------------------------------------------------------------
// Kernel 2: 3x3 conv (pad=1) as implicit GEMM, K = 9 * 64.
// grid = (128 rows, 8 images), block = 256 (8 waves); wave: 64 oc x 16 px.
// Input channels processed in chunks of 16:
//   - the 3 needed input rows are staged into LDS via async b128 copies
//     (halo columns zeroed), so the hot loop never touches global memory
//   - weights are staged pre-swizzled into the exact WMMA A-fragment layout
//     -> lane-consecutive ds_load_b64, conflict-free
// ---------------------------------------------------------------------------
#define ICC  16    // input-channel chunk
#define SROW 136   // LDS floats per staged ic-row (16B-aligned stride)
                   // halo x=-1 at col 3, data x at col 4+x, halo x=128 at col 132

__global__ __launch_bounds__(256)
void conv3x3_kernel(const float* __restrict__ xf,    // (8,64,128,128)
                    const float* __restrict__ w3,    // (64,128,3,3)
                    int icOff,                       // 0: u-half, 64: v-half
                    float* __restrict__ out)         // (8,64,128,128)
{
    __shared__ __attribute__((aligned(16))) float sIn[3 * ICC * SROW];  // 26112 B
    __shared__ float sWf[9 * 4 * 4 * 32 * 2];                           // 36864 B

    const int tid  = threadIdx.x;
    const int wave = tid >> 5;
    const int lane = tid & 31;
    const int ln   = lane & 15;
    const int hi   = lane >> 4;

    const int y    = blockIdx.x;
    const int img  = blockIdx.y;
    const int xcol = wave * 16 + ln;
    const float* xi = xf + (size_t)img * 64 * HW;

    v8f acc[4] = {v8f{}, v8f{}, v8f{}, v8f{}};

    for (int cc = 0; cc < 64; cc += ICC) {
        __syncthreads();   // protect previous chunk's LDS reads

        // ---- stage weights pre-swizzled into WMMA A-fragment layout ----
        // sWf[((e*4 + k04)*4 + m)*64 + L*2 + t] = W3[m*16+(L&15)][cc + k04*4 + (L>>4)*2 + t][e]
        for (int i = tid; i < 9 * 4 * 4 * 32 * 2; i += 256) {
            const int tb  = i & 1;
            const int t   = i >> 1;
            const int L   = t & 31;
            const int m   = (t >> 5) & 3;
            const int k04 = (t >> 7) & 3;
            const int e   = t >> 9;
            const int oc  = m * 16 + (L & 15);
            const int ic  = cc + k04 * 4 + (L >> 4) * 2 + tb;
            sWf[i] = w3[(size_t)(oc * 128 + icOff + ic) * 9 + e];
        }

        // ---- stage input rows y-1..y+1 for this ic chunk (async -> LDS) ----
        for (int i = tid; i < 3 * ICC * 32; i += 256) {
            const int j  = i & 31;       // float4 index within row
            const int t  = i >> 5;
            const int ic = t % ICC;
            const int dy = t / ICC;
            const int yy = y + dy - 1;
            float* dst = &sIn[(dy * ICC + ic) * SROW + 4 + j * 4];
            if (yy >= 0 && yy < 128) {
                const float* src = xi + (size_t)(cc + ic) * HW + (size_t)yy * WID + j * 4;
                copy16_g2l(src, dst);
            }
        }
        // zero halo columns
        if (tid < 3 * ICC) {
            float* row = &sIn[tid * SROW];
            row[3]   = 0.0f;
            row[132] = 0.0f;
        }
        async_wait_all();
        __syncthreads();

        for (int e = 0; e < 9; ++e) {
            const int dy = e / 3;
            const int dx = e - dy * 3;
            const int yy = y + dy - 1;
            if (yy < 0 || yy >= 128) continue;   // uniform across block
            const float* rowB = &sIn[dy * ICC * SROW];
            const float* wE   = &sWf[e * 4 * 4 * 64];

            #pragma unroll
            for (int k0 = 0; k0 < ICC; k0 += 4) {
                const int krL = k0 + hi * 2;
                v2f b;
                b.x = rowB[(size_t)krL       * SROW + 3 + xcol + dx];
                b.y = rowB[(size_t)(krL + 1) * SROW + 3 + xcol + dx];
                const float* wK = wE + (k0 >> 2) * 4 * 64 + lane * 2;
                v2f a0 = *(const v2f*)(wK + 0 * 64);
                v2f a1 = *(const v2f*)(wK + 1 * 64);
                v2f a2 = *(const v2f*)(wK + 2 * 64);
                v2f a3 = *(const v2f*)(wK + 3 * 64);
                acc[0] = __builtin_amdgcn_wmma_f32_16x16x4_f32(false, a0, false, b, (short)0, acc[0], false, false);
                acc[1] = __builtin_amdgcn_wmma_f32_16x16x4_f32(false, a1, false, b, (short)0, acc[1], false, false);
                acc[2] = __builtin_amdgcn_wmma_f32_16x16x4_f32(false, a2, false, b, (short)0, acc[2], false, false);
                acc[3] = __builtin_amdgcn_wmma_f32_16x16x4_f32(false, a3, false, b, (short)0, acc[3], false, false);
            }
        }
    }

    float* oi = out + (size_t)img * 64 * HW + (size_t)y * WID + xcol;
    #pragma unroll
    for (int m = 0; m < 4; ++m) {
        #pragma unroll
        for (int r = 0; r < 8; ++r) {
            const int oc = m * 16 + hi * 8 + r;
            oi[(size_t)oc * HW] = acc[m][r];
        }
    }
}

// ---------------------------------------------------------------------------
// Kernel 3: broadcast add  out[b,sx,sy] = ou[b*4+sx] + ov[b*4+sy] + bias[oc]
// reads are L2-hot; the 128 MB output is write-once -> non-temporal stores
// ---------------------------------------------------------------------------
__global__ __launch_bounds__(256)
void add_broadcast_kernel(const f4* __restrict__ ou,
                          const f4* __restrict__ ov,
                          const float* __restrict__ bias,
                          f4* __restrict__ out,
                          int total4)
{
    const int i = blockIdx.x * 256 + threadIdx.x;
    if (i >= total4) return;
    const int inner = i & (262144 - 1);      // 64*16384/4 per (b,sx,sy)
    const int grp   = i >> 18;               // b*16 + sx*4 + sy
    const int b     = grp >> 4;
    const int sx    = (grp >> 2) & 3;
    const int sy    = grp & 3;
    const int oc    = inner >> 12;

    const f4 a = ou[(size_t)(b * 4 + sx) * 262144 + inner];
    const f4 c = ov[(size_t)(b * 4 + sy) * 262144 + inner];
    const float bi = bias[oc];
    f4 r = a + c + (f4){bi, bi, bi, bi};
    __builtin_nontemporal_store(r, &out[i]);
}

// ---------------------------------------------------------------------------
// Launch: proj(u), proj(v) -> conv3x3(uf), conv3x3(vf) -> broadcast add
// workspace layout: uf | vf | ou | ov, each 8*64*128*128 floats (33.5 MB)
// ---------------------------------------------------------------------------
extern "C" void kernel_launch(void* const* d_in, const int* in_sizes, int n_in,
                              void* d_out, int out_size, void* d_ws, size_t ws_size,
                              hipStream_t stream) {
    const float* u        = (const float*)d_in[0];
    const float* v        = (const float*)d_in[1];
    const float* pu_w     = (const float*)d_in[2];
    const float* pu_gamma = (const float*)d_in[3];
    const float* pu_beta  = (const float*)d_in[4];
    const float* pu_mean  = (const float*)d_in[5];
    const float* pu_var   = (const float*)d_in[6];
    const float* pv_w     = (const float*)d_in[7];
    const float* pv_gamma = (const float*)d_in[8];
    const float* pv_beta  = (const float*)d_in[9];
    const float* pv_mean  = (const float*)d_in[10];
    const float* pv_var   = (const float*)d_in[11];
    const float* conv_w   = (const float*)d_in[12];
    const float* conv_b   = (const float*)d_in[13];

    const size_t imgElems = (size_t)8 * 64 * HW;
    float* uf = (float*)d_ws;
    float* vf = uf + imgElems;
    float* ou = vf + imgElems;
    float* ov = ou + imgElems;

    dim3 blk(256);
    dim3 grd(128, 8);

    proj_bn_relu_kernel<<<grd, blk, 0, stream>>>(u, pu_w, pu_gamma, pu_beta, pu_mean, pu_var, uf);
    proj_bn_relu_kernel<<<grd, blk, 0, stream>>>(v, pv_w, pv_gamma, pv_beta, pv_mean, pv_var, vf);

    conv3x3_kernel<<<grd, blk, 0, stream>>>(uf, conv_w, 0,  ou);
    conv3x3_kernel<<<grd, blk, 0, stream>>>(vf, conv_w, 64, ov);

    const int total4 = (int)((size_t)2 * 4 * 4 * 64 * HW / 4);  // 8,388,608
    add_broadcast_kernel<<<dim3((total4 + 255) / 256), blk, 0, stream>>>(
        (const f4*)ou, (const f4*)ov, conv_b, (f4*)((float*)d_out), total4);
}